// GATHLayer_34059090657352
// MI455X (gfx1250) — compile-verified
//
#include <hip/hip_runtime.h>
#include <math.h>

#define DIM 256
#define HEADS 8
#define HEAD_DIM 32

typedef __attribute__((ext_vector_type(2))) float v2f;
typedef __attribute__((ext_vector_type(8))) float v8f;

// ---------------------------------------------------------------------------
// Fused QKV projection: one wave32 computes the same 16x16 tile of
// Q = H@Wq^T+bq, K = H@Wk^T+bk, V = H@Wv^T+bv, sharing every A fragment
// across three V_WMMA_F32_16X16X4_F32 issues per K-step.
// M must be a multiple of 16 (N=100000 is).
//
// A (16x4 MxK): lanes 0-15 -> M=lane, VGPR0=K0, VGPR1=K1;
//               lanes 16-31 -> M=lane-16, VGPR0=K2, VGPR1=K3.
// B (4x16 KxN) mirrors A with N=lane%16.
// C/D (16x16): VGPR v holds M=v (lanes 0-15) / M=v+8 (lanes 16-31), N=lane%16.
// ---------------------------------------------------------------------------
__global__ void wmma_gemm_qkv(const float* __restrict__ H,
                              const float* __restrict__ Wq, const float* __restrict__ bq,
                              const float* __restrict__ Wk, const float* __restrict__ bk,
                              const float* __restrict__ Wv, const float* __restrict__ bv,
                              float* __restrict__ Q, float* __restrict__ K,
                              float* __restrict__ V, int M, int Kd, int Nd) {
    const int gwave = (int)((blockIdx.x * blockDim.x + threadIdx.x) >> 5);
    const int lane  = (int)(threadIdx.x & 31);
    const int ntN   = Nd >> 4;
    const int rt    = gwave / ntN;
    const int ct    = gwave - rt * ntN;
    if (rt * 16 >= M) return;

    const int row0 = rt * 16;
    const int col0 = ct * 16;
    const int half = lane >> 4;
    const int l    = lane & 15;

    const float* aptr  = H  + (size_t)(row0 + l) * Kd + 2 * half;
    const size_t boff  = (size_t)(col0 + l) * Kd + 2 * half;
    const float* bqptr = Wq + boff;
    const float* bkptr = Wk + boff;
    const float* bvptr = Wv + boff;

    v8f accq = {}, acck = {}, accv = {};
#pragma unroll 8
    for (int k0 = 0; k0 < Kd; k0 += 4) {
        v2f a   = *(const v2f*)(aptr  + k0);
        v2f fbq = *(const v2f*)(bqptr + k0);
        v2f fbk = *(const v2f*)(bkptr + k0);
        v2f fbv = *(const v2f*)(bvptr + k0);
        accq = __builtin_amdgcn_wmma_f32_16x16x4_f32(false, a, false, fbq,
                                                     (short)0, accq, false, false);
        acck = __builtin_amdgcn_wmma_f32_16x16x4_f32(false, a, false, fbk,
                                                     (short)0, acck, false, false);
        accv = __builtin_amdgcn_wmma_f32_16x16x4_f32(false, a, false, fbv,
                                                     (short)0, accv, false, false);
    }

    const float biasq = bq[col0 + l];
    const float biask = bk[col0 + l];
    const float biasv = bv[col0 + l];
#pragma unroll
    for (int v = 0; v < 8; ++v) {
        const size_t idx = (size_t)(row0 + v + 8 * half) * Nd + col0 + l;
        Q[idx] = accq[v] + biasq;
        K[idx] = acck[v] + biask;
        V[idx] = accv[v] + biasv;
    }
}

// ---------------------------------------------------------------------------
// Generic wave-per-tile WMMA GEMM: Y = X @ W^T + bias, optional LeakyReLU(0.2).
// Uniform fast path when the whole 16-row tile is in range (no mask VALU);
// masked path only for the final partial tile (used by the 200-row rel proj).
// ---------------------------------------------------------------------------
__global__ void wmma_gemm_xwT_bias(const float* __restrict__ X,
                                   const float* __restrict__ W,
                                   const float* __restrict__ bias,
                                   float* __restrict__ Y,
                                   int M, int Kd, int Nd, int leaky) {
    const int gwave = (int)((blockIdx.x * blockDim.x + threadIdx.x) >> 5);
    const int lane  = (int)(threadIdx.x & 31);
    const int ntN   = Nd >> 4;
    const int rt    = gwave / ntN;
    const int ct    = gwave - rt * ntN;
    if (rt * 16 >= M) return;

    const int row0 = rt * 16;
    const int col0 = ct * 16;
    const int half = lane >> 4;
    const int l    = lane & 15;

    const float* bptr = W + (size_t)(col0 + l) * Kd + 2 * half;

    v8f acc = {};
    if (row0 + 16 <= M) {
        // full tile: no per-lane masking anywhere in the WMMA loop
        const float* aptr = X + (size_t)(row0 + l) * Kd + 2 * half;
#pragma unroll 8
        for (int k0 = 0; k0 < Kd; k0 += 4) {
            v2f a = *(const v2f*)(aptr + k0);
            v2f b = *(const v2f*)(bptr + k0);
            acc = __builtin_amdgcn_wmma_f32_16x16x4_f32(false, a, false, b,
                                                        (short)0, acc, false, false);
        }
    } else {
        // partial tile: clamp row, zero contribution of padding rows
        const int   arow  = row0 + l;
        const float amask = (arow < M) ? 1.0f : 0.0f;
        const float* aptr = X + (size_t)((arow < M) ? arow : 0) * Kd + 2 * half;
#pragma unroll 4
        for (int k0 = 0; k0 < Kd; k0 += 4) {
            v2f a = *(const v2f*)(aptr + k0);
            v2f b = *(const v2f*)(bptr + k0);
            a *= amask;
            acc = __builtin_amdgcn_wmma_f32_16x16x4_f32(false, a, false, b,
                                                        (short)0, acc, false, false);
        }
    }

    const float bv = bias[col0 + l];
#pragma unroll
    for (int v = 0; v < 8; ++v) {
        const int r = row0 + v + 8 * half;
        if (r < M) {
            float val = acc[v] + bv;
            if (leaky) val = (val >= 0.0f) ? val : 0.2f * val;
            Y[(size_t)r * Nd + col0 + l] = val;
        }
    }
}

// ---------------------------------------------------------------------------
// One wave32 per edge. Head h = chunk h of 32 dims; lane d covers dim h*32+d.
// score[h] = att_b + sum_d att_w[d] * Q[dst][h,d] * (K[src][h,d] + Rp[rel][h,d])
// w = sigmoid(score); msg = w * (V[src] + Rp); atomicAdd into ACC[dst].
// attn_mean[e] = mean_h(w).
// ---------------------------------------------------------------------------
__global__ void edge_attention_scatter(const float* __restrict__ Q,
                                       const float* __restrict__ K,
                                       const float* __restrict__ V,
                                       const float* __restrict__ Rp,
                                       const int* __restrict__ eidx,
                                       const int* __restrict__ etype,
                                       const float* __restrict__ att_w,
                                       const float* __restrict__ att_b,
                                       float* __restrict__ acc,
                                       float* __restrict__ attn_mean,
                                       int E) {
    const int e    = (int)((blockIdx.x * blockDim.x + threadIdx.x) >> 5);
    const int lane = (int)(threadIdx.x & 31);
    if (e >= E) return;

    const int src = eidx[e];
    const int dst = eidx[E + e];
    const int rel = etype[e];

    const float* q = Q  + (size_t)dst * DIM;
    const float* k = K  + (size_t)src * DIM;
    const float* v = V  + (size_t)src * DIM;
    const float* r = Rp + (size_t)rel * DIM;

    const float aw = att_w[lane];
    const float ab = att_b[0];

    float wsum = 0.0f;
#pragma unroll
    for (int h = 0; h < HEADS; ++h) {
        const int d  = h * HEAD_DIM + lane;
        const float rv = r[d];
        float t = q[d] * (k[d] + rv) * aw;
        for (int off = 16; off > 0; off >>= 1)
            t += __shfl_xor(t, off, 32);
        const float s = t + ab;
        const float w = 1.0f / (1.0f + __expf(-s));
        const float msg = w * (v[d] + rv);
        atomicAdd(&acc[(size_t)dst * DIM + d], msg);
        wsum += w;
    }
    if (lane == 0) attn_mean[e] = wsum * (1.0f / HEADS);
}

// ---------------------------------------------------------------------------
// One wave32 per node row: H_out = LayerNorm(H + O) * g + b  (eps = 1e-5).
// ---------------------------------------------------------------------------
__global__ void residual_layernorm(const float* __restrict__ H,
                                   const float* __restrict__ O,
                                   const float* __restrict__ g,
                                   const float* __restrict__ b,
                                   float* __restrict__ out,
                                   int N) {
    const int row  = (int)((blockIdx.x * blockDim.x + threadIdx.x) >> 5);
    const int lane = (int)(threadIdx.x & 31);
    if (row >= N) return;

    const float* h = H + (size_t)row * DIM;
    const float* o = O + (size_t)row * DIM;

    float x[8];
    float sum = 0.0f;
#pragma unroll
    for (int i = 0; i < 8; ++i) {
        const int d = i * 32 + lane;
        x[i] = h[d] + o[d];
        sum += x[i];
    }
    for (int off = 16; off > 0; off >>= 1)
        sum += __shfl_xor(sum, off, 32);
    const float mu = sum * (1.0f / DIM);

    float vs = 0.0f;
#pragma unroll
    for (int i = 0; i < 8; ++i) {
        const float dd = x[i] - mu;
        vs += dd * dd;
    }
    for (int off = 16; off > 0; off >>= 1)
        vs += __shfl_xor(vs, off, 32);
    const float inv = rsqrtf(vs * (1.0f / DIM) + 1e-5f);

#pragma unroll
    for (int i = 0; i < 8; ++i) {
        const int d = i * 32 + lane;
        out[(size_t)row * DIM + d] = (x[i] - mu) * inv * g[d] + b[d];
    }
}

extern "C" void kernel_launch(void* const* d_in, const int* in_sizes, int n_in,
                              void* d_out, int out_size, void* d_ws, size_t ws_size,
                              hipStream_t stream) {
    const float* H      = (const float*)d_in[0];
    const float* R      = (const float*)d_in[1];
    const float* Wq_w   = (const float*)d_in[2];
    const float* Wq_b   = (const float*)d_in[3];
    const float* Wk_w   = (const float*)d_in[4];
    const float* Wk_b   = (const float*)d_in[5];
    const float* Wv_w   = (const float*)d_in[6];
    const float* Wv_b   = (const float*)d_in[7];
    const float* Wrel_w = (const float*)d_in[8];
    const float* Wrel_b = (const float*)d_in[9];
    const float* att_w  = (const float*)d_in[10];
    const float* att_b  = (const float*)d_in[11];
    const float* out_w  = (const float*)d_in[12];
    const float* out_b  = (const float*)d_in[13];
    const float* ln_g   = (const float*)d_in[14];
    const float* ln_b   = (const float*)d_in[15];
    const int*   eidx   = (const int*)d_in[16];
    const int*   etype  = (const int*)d_in[17];

    const int N     = in_sizes[0] / DIM;   // 100000 (multiple of 16)
    const int NRELS = in_sizes[1] / DIM;   // 200
    const int E     = in_sizes[17];        // 300000

    // Workspace layout (floats): Q | K | V | ACC | Rp.  O reuses Q region.
    float* Q   = (float*)d_ws;
    float* Kp  = Q   + (size_t)N * DIM;
    float* Vp  = Kp  + (size_t)N * DIM;
    float* ACC = Vp  + (size_t)N * DIM;
    float* Rp  = ACC + (size_t)N * DIM;
    float* O   = Q;  // Q/K/V dead after edge pass

    float* Hout      = (float*)d_out;
    float* attn_mean = Hout + (size_t)N * DIM;

    // Zero scatter accumulator (deterministic per launch; capture-safe).
    hipMemsetAsync(ACC, 0, (size_t)N * DIM * sizeof(float), stream);

    const int ntN = DIM / 16;  // 16 column tiles

    // --- Fused Q/K/V projections (WMMA, shared A fragments) ---
    {
        const int waves  = (N / 16) * ntN;          // 100000 waves
        const int blocks = (waves * 32 + 255) / 256;
        wmma_gemm_qkv<<<blocks, 256, 0, stream>>>(H, Wq_w, Wq_b, Wk_w, Wk_b, Wv_w, Wv_b,
                                                  Q, Kp, Vp, N, DIM, DIM);
    }

    // --- Relation embedding projection (tiny, has one partial tile) ---
    {
        const int waves  = ((NRELS + 15) / 16) * ntN;
        const int blocks = (waves * 32 + 255) / 256;
        wmma_gemm_xwT_bias<<<blocks, 256, 0, stream>>>(R, Wrel_w, Wrel_b, Rp,
                                                       NRELS, DIM, DIM, 0);
    }

    // --- Edge gather / score / sigmoid / scatter-add ---
    {
        const int blocks = (E * 32 + 255) / 256;    // one wave per edge
        edge_attention_scatter<<<blocks, 256, 0, stream>>>(Q, Kp, Vp, Rp, eidx, etype,
                                                           att_w, att_b, ACC, attn_mean, E);
    }

    // --- Output projection + LeakyReLU(0.2) (into O = old Q region) ---
    {
        const int waves  = (N / 16) * ntN;
        const int blocks = (waves * 32 + 255) / 256;
        wmma_gemm_xwT_bias<<<blocks, 256, 0, stream>>>(ACC, out_w, out_b, O,
                                                       N, DIM, DIM, 1);
    }

    // --- Residual + LayerNorm ---
    {
        const int blocks = (N * 32 + 255) / 256;    // one wave per row
        residual_layernorm<<<blocks, 256, 0, stream>>>(H, O, ln_g, ln_b, Hout, N);
    }
}